// Freq_Att_74285754352014
// MI455X (gfx1250) — compile-verified
//
#include <hip/hip_runtime.h>

// ---------------------------------------------------------------------------
// Freq-Att on MI455X (gfx1250): bandwidth-bound restructure.
//   K1: per-(b,c) channel means (1 full read, 201 MB)
//   K2: all three SE gates collapsed to G[b,c]; effective 3x3 weights
//       W_eff[b, src_channel, tap] = G * sp_w  (tiny)
//   K3: WMMA fp32 tap-GEMM  U[b,tap,p] = sum_c x[b,c,p] * W_eff[b,c,tap]
//       (V_WMMA_F32_16X16X4_F32, M=16 pixels, N=16 taps (9 used), K=192)
//   K4: 9-point stencil -> s_attn = |conv|
//   K5: out[b,c',p] = x[b,perm(c'),p] * G[b,c'] * s_attn[b,p]  (stream)
// ---------------------------------------------------------------------------

typedef __attribute__((ext_vector_type(2))) float v2f;
typedef __attribute__((ext_vector_type(8))) float v8f;

#define NB   64
#define NC   192
#define HW   4096
#define WW   64

__constant__ int c_idx_low[32] = {
  0,1,8,16,9,2,3,10,17,24,32,25,18,11,4,5,
  12,19,26,33,40,48,41,34,27,20,13,6,7,14,21,28};
__constant__ int c_idx_high[32] = {
  35,42,49,56,57,50,43,36,29,22,15,23,30,37,44,51,
  58,59,52,45,38,31,39,46,53,60,61,54,47,55,62,63};

// feat_DCT channel c' -> source channel of DCT_x
__device__ __forceinline__ int perm_src(int cp) {
  int comp = cp >> 6;
  int t    = cp & 63;
  int idx  = (t < 32) ? c_idx_low[t] : c_idx_high[t - 32];
  return comp * 64 + idx;
}

// ---------------- K1: per-(b,c) means -------------------------------------
__global__ void k_mean(const float* __restrict__ x, float* __restrict__ mean) {
  int blk = blockIdx.x;                       // b*192 + c
  const float4* x4 = (const float4*)(x + (size_t)blk * HW);
  int tid = threadIdx.x;
  float s = 0.f;
#pragma unroll
  for (int k = 0; k < 4; ++k) {               // 4096 floats / 256 thr / 4 = 4
    float4 v = x4[tid + k * 256];
    s += v.x + v.y + v.z + v.w;
  }
  __shared__ float red[256];
  red[tid] = s;
  __syncthreads();
  for (int off = 128; off > 0; off >>= 1) {
    if (tid < off) red[tid] += red[tid + off];
    __syncthreads();
  }
  if (tid == 0) mean[blk] = red[0] * (1.0f / HW);
}

// ---------------- K2: gates -----------------------------------------------
// one block per batch, 192 threads
__global__ void k_gates(const float* __restrict__ mean,
                        const float* __restrict__ hb_w1, const float* __restrict__ hb_b1,
                        const float* __restrict__ hb_w2, const float* __restrict__ hb_b2,
                        const float* __restrict__ lb_w1, const float* __restrict__ lb_b1,
                        const float* __restrict__ lb_w2, const float* __restrict__ lb_b2,
                        const float* __restrict__ b_w1,  const float* __restrict__ b_b1,
                        const float* __restrict__ b_w2,  const float* __restrict__ b_b2,
                        const float* __restrict__ sp_w,
                        float* __restrict__ G, float* __restrict__ weff) {
  int b = blockIdx.x;
  int t = threadIdx.x;                        // 0..191
  __shared__ float sm[NC], smh[96], sml[96], sh1[6], sl1[6],
                   sgh[96], sgl[96], smfd[NC], sh2[12];
  sm[t] = mean[b * NC + t];
  __syncthreads();
  if (t < 96) {                               // gather high/low means
    int comp = t >> 5, i = t & 31;
    smh[t] = sm[comp * 64 + c_idx_high[i]];
    sml[t] = sm[comp * 64 + c_idx_low[i]];
  }
  __syncthreads();
  if (t < 6) {                                // 96 -> 6, relu
    float a = hb_b1[t], c = lb_b1[t];
    for (int j = 0; j < 96; ++j) {
      a += hb_w1[t * 96 + j] * smh[j];
      c += lb_w1[t * 96 + j] * sml[j];
    }
    sh1[t] = fmaxf(a, 0.f);
    sl1[t] = fmaxf(c, 0.f);
  }
  __syncthreads();
  if (t < 96) {                               // 6 -> 96, abs
    float a = hb_b2[t], c = lb_b2[t];
    for (int k = 0; k < 6; ++k) {
      a += hb_w2[t * 6 + k] * sh1[k];
      c += lb_w2[t * 6 + k] * sl1[k];
    }
    sgh[t] = fabsf(a);
    sgl[t] = fabsf(c);
  }
  __syncthreads();
  // feat_DCT layout: per comp, first 32 = low, next 32 = high
  int comp = t >> 6, tt = t & 63;
  float g1 = (tt < 32) ? sgl[comp * 32 + tt] : sgh[comp * 32 + tt - 32];
  int src = perm_src(t);
  smfd[t] = sm[src] * g1;                     // pooled mean of gated feat_DCT
  __syncthreads();
  if (t < 12) {                               // 192 -> 12, relu
    float a = b_b1[t];
    for (int j = 0; j < NC; ++j) a += b_w1[t * NC + j] * smfd[j];
    sh2[t] = fmaxf(a, 0.f);
  }
  __syncthreads();
  float g2 = b_b2[t];                         // 12 -> 192, abs
  for (int k = 0; k < 12; ++k) g2 += b_w2[t * 12 + k] * sh2[k];
  g2 = fabsf(g2);
  float g = g1 * g2;
  G[b * NC + t] = g;
  // effective conv weights indexed by SOURCE channel (natural x layout)
  float* wdst = weff + ((size_t)b * NC + src) * 16;
  for (int tap = 0; tap < 9; ++tap) wdst[tap] = g * sp_w[t * 9 + tap];
  for (int tap = 9; tap < 16; ++tap) wdst[tap] = 0.f;
}

// ---------------- K3: WMMA fp32 tap-GEMM ----------------------------------
// U[b, tap, p] = sum_c x[b,c,p] * weff[b,c,tap]
// block: 256 thr = 8 waves; each wave: 4 pixel-tiles of 16 -> 64 pixels
__global__ void k_tapgemm(const float* __restrict__ x,
                          const float* __restrict__ weff,
                          float* __restrict__ U) {
  __shared__ float lw[NC * 16];               // 12 KB: per-batch W_eff
  int b    = blockIdx.x >> 3;
  int tile = blockIdx.x & 7;                  // 512 pixels per block
  for (int i = threadIdx.x; i < NC * 16; i += 256)
    lw[i] = weff[(size_t)b * NC * 16 + i];
  __syncthreads();

  int lane = threadIdx.x & 31;
  int wid  = threadIdx.x >> 5;
  int pixBase = tile * 512 + wid * 64;
  const float* xb = x + (size_t)b * NC * HW;

  int m     = lane & 15;                      // A: M index / B,C: N index
  int khalf = (lane >> 4) << 1;               // 0 (lanes 0-15) or 2 (16-31)

  v8f acc[4] = {};
  for (int k0 = 0; k0 < NC; k0 += 4) {
    int c0 = k0 + khalf;
    v2f Bf;                                   // B: 4x16 (K x taps)
    Bf.x = lw[c0 * 16 + m];
    Bf.y = lw[(c0 + 1) * 16 + m];
    const float* xc = xb + (size_t)c0 * HW;
#pragma unroll
    for (int t = 0; t < 4; ++t) {
      int p = pixBase + t * 16 + m;
      v2f Af;                                 // A: 16x4 (pixels x K)
      Af.x = xc[p];
      Af.y = xc[HW + p];
      acc[t] = __builtin_amdgcn_wmma_f32_16x16x4_f32(
          false, Af, false, Bf, (short)0, acc[t], false, false);
    }
  }
  // C layout: lane -> N(tap), VGPR v -> M = (lane<16?0:8)+v
  int tap   = lane & 15;
  int mbase = (lane >> 4) << 3;
  if (tap < 9) {
    float* Ub = U + ((size_t)b * 16 + tap) * HW + pixBase + mbase;
#pragma unroll
    for (int t = 0; t < 4; ++t)
#pragma unroll
      for (int v = 0; v < 8; ++v) Ub[t * 16 + v] = acc[t][v];
  }
}

// ---------------- K4: 9-point stencil -> s_attn ---------------------------
__global__ void k_sattn(const float* __restrict__ U,
                        const float* __restrict__ sp_b,
                        float* __restrict__ sattn) {
  int idx = blockIdx.x * 256 + threadIdx.x;   // over 64*4096
  int b = idx >> 12, p = idx & 4095;
  int h = p >> 6, w = p & 63;
  float s = sp_b[0];
#pragma unroll
  for (int t = 0; t < 9; ++t) {
    int hh = h + t / 3 - 1, ww = w + t % 3 - 1;
    if (hh >= 0 && hh < WW && ww >= 0 && ww < WW)
      s += U[((size_t)b * 16 + t) * HW + hh * WW + ww];
  }
  sattn[idx] = fabsf(s);
}

// ---------------- K5: final streamed output -------------------------------
__global__ void k_out(const float* __restrict__ x, const float* __restrict__ G,
                      const float* __restrict__ sattn, float* __restrict__ out) {
  int n = blockIdx.x * 256 + threadIdx.x;     // float4 groups: 64*192*1024
  int b  = n / (NC * 1024);
  int r  = n - b * (NC * 1024);
  int cp = r >> 10;
  int p4 = r & 1023;
  int src = perm_src(cp);
  float4 v = ((const float4*)(x + ((size_t)b * NC + src) * HW))[p4];
  float  g = G[b * NC + cp];
  float4 s = ((const float4*)(sattn + (size_t)b * HW))[p4];
  float4 o;
  o.x = v.x * g * s.x; o.y = v.y * g * s.y;
  o.z = v.z * g * s.z; o.w = v.w * g * s.w;
  ((float4*)out)[n] = o;
}

// ---------------------------------------------------------------------------
extern "C" void kernel_launch(void* const* d_in, const int* in_sizes, int n_in,
                              void* d_out, int out_size, void* d_ws, size_t ws_size,
                              hipStream_t stream) {
  (void)in_sizes; (void)n_in; (void)out_size; (void)ws_size;
  const float* x     = (const float*)d_in[0];
  const float* hb_w1 = (const float*)d_in[1];
  const float* hb_b1 = (const float*)d_in[2];
  const float* hb_w2 = (const float*)d_in[3];
  const float* hb_b2 = (const float*)d_in[4];
  const float* lb_w1 = (const float*)d_in[5];
  const float* lb_b1 = (const float*)d_in[6];
  const float* lb_w2 = (const float*)d_in[7];
  const float* lb_b2 = (const float*)d_in[8];
  const float* b_w1  = (const float*)d_in[9];
  const float* b_b1  = (const float*)d_in[10];
  const float* b_w2  = (const float*)d_in[11];
  const float* b_b2  = (const float*)d_in[12];
  const float* sp_w  = (const float*)d_in[13];
  const float* sp_b  = (const float*)d_in[14];
  float* out = (float*)d_out;

  float* ws    = (float*)d_ws;                 // fp32 workspace layout:
  float* mean  = ws;                           // 64*192        = 12288
  float* G     = ws + 12288;                   // 64*192        = 12288
  float* weff  = ws + 24576;                   // 64*192*16     = 196608
  float* U     = ws + 221184;                  // 64*16*4096    = 4194304
  float* sattn = ws + 4415488;                 // 64*4096       = 262144
                                               // total ~17.9 MB

  k_mean   <<<NB * NC, 256, 0, stream>>>(x, mean);
  k_gates  <<<NB, NC, 0, stream>>>(mean, hb_w1, hb_b1, hb_w2, hb_b2,
                                   lb_w1, lb_b1, lb_w2, lb_b2,
                                   b_w1, b_b1, b_w2, b_b2, sp_w, G, weff);
  k_tapgemm<<<NB * 8, 256, 0, stream>>>(x, weff, U);
  k_sattn  <<<(NB * HW) / 256, 256, 0, stream>>>(U, sp_b, sattn);
  k_out    <<<(NB * NC * HW / 4) / 256, 256, 0, stream>>>(x, G, sattn, out);
}